// DivExpert_62380105007399
// MI455X (gfx1250) — compile-verified
//
#include <hip/hip_runtime.h>
#include <stdint.h>

// DivExpert reduced to exact u32 arithmetic (see analysis):
//   per row: a,b from nibbles; 8x wrapping-u32 Newton-Raphson reciprocal;
//   quotient + 20-step remainder correction; write nibbles(q) to cols 16..23;
//   all other 200 columns are a straight copy of x.
// Memory-bound (~218 MB total @ 23.3 TB/s ~ 9.4us). Fused single pass staged
// through LDS using CDNA5 async global<->LDS B128 ops (ASYNCcnt).

#define DIM            208
#define ROWS_PER_BLOCK 16
#define CHUNKS_PER_ROW (DIM / 4)   // 52 float4 per row
#define BLOCK          256

__device__ __forceinline__ void async_load_b128(uint32_t lds_off, uint64_t gaddr) {
    // GLOBAL_LOAD_ASYNC_TO_LDS_B128: vdst = LDS byte offset VGPR, vaddr = 64-bit global addr
    asm volatile("global_load_async_to_lds_b128 %0, %1, off"
                 :: "v"(lds_off), "v"(gaddr) : "memory");
}

__device__ __forceinline__ void async_store_b128(uint64_t gaddr, uint32_t lds_off) {
    // GLOBAL_STORE_ASYNC_FROM_LDS_B128: vaddr = 64-bit global addr, vsrc = LDS byte offset VGPR
    asm volatile("global_store_async_from_lds_b128 %0, %1, off"
                 :: "v"(gaddr), "v"(lds_off) : "memory");
}

__device__ __forceinline__ void wait_async0() {
    asm volatile("s_wait_asynccnt 0" ::: "memory");
}

__global__ void __launch_bounds__(BLOCK)
div_expert_kernel(const float* __restrict__ x, float* __restrict__ out, int nrows) {
    __shared__ __align__(16) float tile[ROWS_PER_BLOCK * DIM];   // 13,312 bytes

    const int tid     = threadIdx.x;
    const int rowBase = blockIdx.x * ROWS_PER_BLOCK;
    int rowsHere = nrows - rowBase;
    if (rowsHere <= 0) return;
    if (rowsHere > ROWS_PER_BLOCK) rowsHere = ROWS_PER_BLOCK;
    const int nchunks = rowsHere * CHUNKS_PER_ROW;

    // LDS flat address low 32 bits == workgroup-relative LDS byte offset
    const uint32_t ldsBase = (uint32_t)(uintptr_t)(&tile[0]);
    const uint64_t srcBase = (uint64_t)(uintptr_t)(x   + (size_t)rowBase * DIM);
    const uint64_t dstBase = (uint64_t)(uintptr_t)(out + (size_t)rowBase * DIM);

    // ---- stage 1: async copy global -> LDS (16 B per chunk, fully coalesced)
    for (int c = tid; c < nchunks; c += BLOCK)
        async_load_b128(ldsBase + (uint32_t)c * 16u, srcBase + (uint64_t)c * 16u);
    wait_async0();
    __syncthreads();

    // ---- stage 2: one thread per row, exact u32 Newton-Raphson division
    if (tid < rowsHere) {
        float* rp = &tile[tid * DIM];
        uint32_t a = 0u, bv = 0u;
        #pragma unroll
        for (int k = 0; k < 8; ++k) {
            a  |= ((uint32_t)rp[0 + k]) << (4 * k);   // nibbles are exact small floats
            bv |= ((uint32_t)rp[8 + k]) << (4 * k);
        }
        const uint32_t b = bv ? bv : 1u;              // max(b, 1)

        uint32_t r = 65536u / b;                      // NR seed (no clamp, per reference)
        #pragma unroll
        for (int it = 0; it < 8; ++it) {
            uint32_t bx   = b * r;                    // (b*r) mod 2^32
            uint32_t corr = 131072u - bx;             // (2*NR_SCALE - bx) mod 2^32
            uint32_t xc   = r * corr;                 // mod 2^32
            uint32_t rn   = xc >> 16;                 // floor div, xc >= 0
            r = rn ? rn : 1u;                         // max(, 1)
        }

        uint32_t q   = (a * r) >> 16;                 // (a*r mod 2^32) // 65536
        uint32_t rem = a - q * b;                     // (a - qb) mod 2^32
        #pragma unroll
        for (int it = 0; it < 20; ++it) {             // rem<0 branch provably dead
            if (rem >= b) { q += 1u; rem -= b; }
        }

        #pragma unroll
        for (int k = 0; k < 8; ++k)
            rp[16 + k] = (float)((q >> (4 * k)) & 15u);  // to_nibbles(q) into RES
    }
    __syncthreads();

    // ---- stage 3: async copy LDS -> global
    for (int c = tid; c < nchunks; c += BLOCK)
        async_store_b128(dstBase + (uint64_t)c * 16u, ldsBase + (uint32_t)c * 16u);
    wait_async0();
}

extern "C" void kernel_launch(void* const* d_in, const int* in_sizes, int n_in,
                              void* d_out, int out_size, void* d_ws, size_t ws_size,
                              hipStream_t stream) {
    const float* x   = (const float*)d_in[0];          // [N, 208] fp32
    float*       out = (float*)d_out;                  // [N, 208] fp32
    const int nrows  = in_sizes[0] / DIM;
    const int grid   = (nrows + ROWS_PER_BLOCK - 1) / ROWS_PER_BLOCK;
    div_expert_kernel<<<grid, BLOCK, 0, stream>>>(x, out, nrows);
}